// GAT_52209622450323
// MI455X (gfx1250) — compile-verified
//
#include <hip/hip_runtime.h>
#include <hip/hip_bf16.h>
#include <cstddef>

// ---------------------------------------------------------------------------
// Problem constants (match reference setup_inputs)
// ---------------------------------------------------------------------------
#define NW   80000
#define ND   16000
#define EWW  200000
#define EWD  200000
#define HH   4
#define FOUT 128
#define NOUT 512   // H * FOUT

typedef float v2f __attribute__((ext_vector_type(2)));
typedef float v8f __attribute__((ext_vector_type(8)));

// ---------------------------------------------------------------------------
// FP32 WMMA GEMM:  C[M, 512] = A[M, K] * B[K, 512]
// One wave32 computes a 16x64 strip of C (4 accumulators) via
// V_WMMA_F32_16X16X4_F32, reusing each A fragment for 4 WMMAs.
// A-frag layout (32-bit A 16x4): lanes 0-15 -> M=0..15, VGPR0 holds K=0 (K=2
// for lanes 16-31), VGPR1 holds K=1 (K=3).  B/C/D: row striped across lanes.
// ---------------------------------------------------------------------------
__global__ __launch_bounds__(256) void gemm_f32_wmma(
    const float* __restrict__ A, const float* __restrict__ B,
    float* __restrict__ C, int M, int K)
{
    const int ngrpN = NOUT / 64;                         // 8 strips of 64 cols
    int wave  = blockIdx.x * (blockDim.x >> 5) + (threadIdx.x >> 5);
    int lane  = threadIdx.x & 31;
    int strips = ((M + 15) >> 4) * ngrpN;
    if (wave >= strips) return;

    int tm = wave / ngrpN, tn = wave % ngrpN;
    int m0 = tm * 16, n0 = tn * 64;
    int half = lane >> 4;        // 0: K pair {0,1}, 1: K pair {2,3}
    int r    = lane & 15;

    int arow = m0 + r;
    if (arow >= M) arow = M - 1;                         // safe clamp
    const float* __restrict__ ap = A + (size_t)arow * K;

    v8f acc0 = {}, acc1 = {}, acc2 = {}, acc3 = {};
    #pragma unroll 2
    for (int k0 = 0; k0 < K; k0 += 4) {
        int ka = k0 + 2 * half;
        v2f a;
        a.x = ap[ka];
        a.y = ap[ka + 1];
        const float* b0 = B + (size_t)ka * NOUT + n0 + r;
        const float* b1 = b0 + NOUT;
        v2f bb0, bb1, bb2, bb3;
        bb0.x = b0[0];  bb0.y = b1[0];
        bb1.x = b0[16]; bb1.y = b1[16];
        bb2.x = b0[32]; bb2.y = b1[32];
        bb3.x = b0[48]; bb3.y = b1[48];
        acc0 = __builtin_amdgcn_wmma_f32_16x16x4_f32(
                   false, a, false, bb0, (short)0, acc0, false, false);
        acc1 = __builtin_amdgcn_wmma_f32_16x16x4_f32(
                   false, a, false, bb1, (short)0, acc1, false, false);
        acc2 = __builtin_amdgcn_wmma_f32_16x16x4_f32(
                   false, a, false, bb2, (short)0, acc2, false, false);
        acc3 = __builtin_amdgcn_wmma_f32_16x16x4_f32(
                   false, a, false, bb3, (short)0, acc3, false, false);
    }
    // D layout: VGPR i, half h -> row m0 + i + 8*h, col (n0 + 16*j) + r
    #pragma unroll
    for (int i = 0; i < 8; ++i) {
        int mr = m0 + i + 8 * half;
        if (mr < M) {
            float* cp = C + (size_t)mr * NOUT + n0 + r;
            cp[0]  = acc0[i];
            cp[16] = acc1[i];
            cp[32] = acc2[i];
            cp[48] = acc3[i];
        }
    }
}

// ---------------------------------------------------------------------------
// el[n,h] = dot(Z[n, h*128 : h*128+128], w[h, :])   (wave per (n,h))
// ---------------------------------------------------------------------------
__global__ __launch_bounds__(256) void head_score(
    const float* __restrict__ Z, const float* __restrict__ w,
    float* __restrict__ out, int N)
{
    int wave = blockIdx.x * 8 + (threadIdx.x >> 5);
    int lane = threadIdx.x & 31;
    if (wave >= N * HH) return;
    int n = wave >> 2, h = wave & 3;

    const float4* zp = (const float4*)(Z + (size_t)n * NOUT + h * FOUT);
    const float4* wp = (const float4*)(w + h * FOUT);
    float4 z = zp[lane], a = wp[lane];
    float s = z.x * a.x + z.y * a.y + z.z * a.z + z.w * a.w;
    #pragma unroll
    for (int off = 16; off > 0; off >>= 1) s += __shfl_down(s, off, 32);
    if (lane == 0) out[n * HH + h] = s;
}

// ---------------------------------------------------------------------------
// Order-preserving float<->uint encoding for atomicMax over signed floats
// ---------------------------------------------------------------------------
__device__ __forceinline__ unsigned f2ord(float f) {
    unsigned u = __float_as_uint(f);
    return (u >> 31) ? ~u : (u | 0x80000000u);
}
__device__ __forceinline__ float ord2f(unsigned u) {
    return (u >> 31) ? __uint_as_float(~u) : __uint_as_float(u & 0x7fffffffu);
}
#define ORD_NEG_INF 0x007FFFFFu   // f2ord(-inf)

__global__ void fill_u32(unsigned* __restrict__ p, unsigned v, int n) {
    int i = blockIdx.x * blockDim.x + threadIdx.x;
    if (i < n) p[i] = v;
}

__device__ __forceinline__ float lrelu(float x) { return x > 0.f ? x : 0.2f * x; }

// pass 1: e = leaky_relu(el[src]+er[dst]); stash e; segment max via atomicMax
__global__ void edge_max(const int* __restrict__ src, const int* __restrict__ dst,
                         const float* __restrict__ el, const float* __restrict__ er,
                         float* __restrict__ ebuf, unsigned* __restrict__ m_u, int E)
{
    int e = blockIdx.x * blockDim.x + threadIdx.x;
    if (e >= E) return;
    int s = src[e], d = dst[e];
    float4 l = *(const float4*)(el + (size_t)s * HH);
    float4 rr = *(const float4*)(er + (size_t)d * HH);
    float4 v;
    v.x = lrelu(l.x + rr.x); v.y = lrelu(l.y + rr.y);
    v.z = lrelu(l.z + rr.z); v.w = lrelu(l.w + rr.w);
    *(float4*)(ebuf + (size_t)e * HH) = v;
    unsigned* mp = m_u + (size_t)d * HH;
    atomicMax(mp + 0, f2ord(v.x)); atomicMax(mp + 1, f2ord(v.y));
    atomicMax(mp + 2, f2ord(v.z)); atomicMax(mp + 3, f2ord(v.w));
}

// pass 2: ex = exp(e - m[dst]); stash ex; segment sum via atomicAdd
__global__ void edge_expsum(const int* __restrict__ dst, float* __restrict__ ebuf,
                            const unsigned* __restrict__ m_u,
                            float* __restrict__ ssum, int E)
{
    int e = blockIdx.x * blockDim.x + threadIdx.x;
    if (e >= E) return;
    int d = dst[e];
    const unsigned* mp = m_u + (size_t)d * HH;
    float4 v = *(float4*)(ebuf + (size_t)e * HH);
    float4 ex;
    ex.x = __expf(v.x - ord2f(mp[0])); ex.y = __expf(v.y - ord2f(mp[1]));
    ex.z = __expf(v.z - ord2f(mp[2])); ex.w = __expf(v.w - ord2f(mp[3]));
    *(float4*)(ebuf + (size_t)e * HH) = ex;
    float* sp = ssum + (size_t)d * HH;
    atomicAdd(sp + 0, ex.x); atomicAdd(sp + 1, ex.y);
    atomicAdd(sp + 2, ex.z); atomicAdd(sp + 3, ex.w);
}

// pass 3: h[dst, f] += sum_h alpha[e,h] * Z[src, h*128 + f]  (wave per edge,
// head-sum fused so scatter target is only [Nd, 128])
__global__ __launch_bounds__(256) void edge_aggregate(
    const int* __restrict__ src, const int* __restrict__ dst,
    const float* __restrict__ ebuf, const float* __restrict__ ssum,
    const float* __restrict__ Zs, float* __restrict__ hbuf, int E)
{
    int e = blockIdx.x * 8 + (threadIdx.x >> 5);
    int lane = threadIdx.x & 31;
    if (e >= E) return;
    int s = src[e], d = dst[e];
    float4 ex = *(const float4*)(ebuf + (size_t)e * HH);
    float4 sv = *(const float4*)(ssum + (size_t)d * HH);
    float a0 = ex.x / sv.x, a1 = ex.y / sv.y, a2 = ex.z / sv.z, a3 = ex.w / sv.w;

    const float4* z = (const float4*)(Zs + (size_t)s * NOUT);
    float4 z0 = z[lane], z1 = z[32 + lane], z2 = z[64 + lane], z3 = z[96 + lane];
    float4 acc;
    acc.x = a0 * z0.x + a1 * z1.x + a2 * z2.x + a3 * z3.x;
    acc.y = a0 * z0.y + a1 * z1.y + a2 * z2.y + a3 * z3.y;
    acc.z = a0 * z0.z + a1 * z1.z + a2 * z2.z + a3 * z3.z;
    acc.w = a0 * z0.w + a1 * z1.w + a2 * z2.w + a3 * z3.w;
    float* hp = hbuf + (size_t)d * FOUT + lane * 4;
    atomicAdd(hp + 0, acc.x); atomicAdd(hp + 1, acc.y);
    atomicAdd(hp + 2, acc.z); atomicAdd(hp + 3, acc.w);
}

// out[n,f] = relu(h[n,f] + sum_h b[h*128+f])
__global__ void bias_relu(const float* __restrict__ hbuf, const float* __restrict__ b,
                          float* __restrict__ out, int N)
{
    int i = blockIdx.x * blockDim.x + threadIdx.x;
    if (i >= N * FOUT) return;
    int f = i & (FOUT - 1);
    float bs = b[f] + b[FOUT + f] + b[2 * FOUT + f] + b[3 * FOUT + f];
    float v = hbuf[i] + bs;
    out[i] = v > 0.f ? v : 0.f;
}

// ---------------------------------------------------------------------------
// Host-side orchestration
// ---------------------------------------------------------------------------
static inline int ceil_div(int a, int b) { return (a + b - 1) / b; }

struct Scratch {
    float *Zs, *Zd, *el, *er, *ssum, *ebuf, *hw, *hd, *xw1, *xd1;
    unsigned* m_u;
};

static void run_relation(const float* Xs, const float* Xd, int fin,
                         const float* W, const float* al, const float* ar,
                         const int* src, const int* dst,
                         int Ns, int Nd, int E,
                         const Scratch& S, float* hbuf, hipStream_t stream)
{
    // projections (fp32 WMMA, 16x64 strip per wave)
    {
        int strips = (Ns / 16) * (NOUT / 64);
        gemm_f32_wmma<<<ceil_div(strips, 8), 256, 0, stream>>>(Xs, W, S.Zs, Ns, fin);
    }
    const float* Zdst = S.Zs;
    if (Xd != Xs) {
        int strips = (Nd / 16) * (NOUT / 64);
        gemm_f32_wmma<<<ceil_div(strips, 8), 256, 0, stream>>>(Xd, W, S.Zd, Nd, fin);
        Zdst = S.Zd;
    }
    // attention scores
    head_score<<<ceil_div(Ns * HH, 8), 256, 0, stream>>>(S.Zs, al, S.el, Ns);
    head_score<<<ceil_div(Nd * HH, 8), 256, 0, stream>>>(Zdst, ar, S.er, Nd);
    // segment-softmax state init
    fill_u32<<<ceil_div(Nd * HH, 256), 256, 0, stream>>>(S.m_u, ORD_NEG_INF, Nd * HH);
    fill_u32<<<ceil_div(Nd * HH, 256), 256, 0, stream>>>((unsigned*)S.ssum, 0u, Nd * HH);
    fill_u32<<<ceil_div(Nd * FOUT, 256), 256, 0, stream>>>((unsigned*)hbuf, 0u, Nd * FOUT);
    // softmax + aggregate
    edge_max   <<<ceil_div(E, 256), 256, 0, stream>>>(src, dst, S.el, S.er, S.ebuf, S.m_u, E);
    edge_expsum<<<ceil_div(E, 256), 256, 0, stream>>>(dst, S.ebuf, S.m_u, S.ssum, E);
    edge_aggregate<<<ceil_div(E, 8), 256, 0, stream>>>(src, dst, S.ebuf, S.ssum, S.Zs, hbuf, E);
}

extern "C" void kernel_launch(void* const* d_in, const int* in_sizes, int n_in,
                              void* d_out, int out_size, void* d_ws, size_t ws_size,
                              hipStream_t stream)
{
    (void)in_sizes; (void)n_in; (void)out_size; (void)ws_size;

    const float* x_word = (const float*)d_in[0];
    const float* x_doc  = (const float*)d_in[1];
    const float* W_ww0  = (const float*)d_in[2];
    const float* al_ww0 = (const float*)d_in[3];
    const float* ar_ww0 = (const float*)d_in[4];
    const float* b_ww0  = (const float*)d_in[5];
    const float* W_wd0  = (const float*)d_in[6];
    const float* al_wd0 = (const float*)d_in[7];
    const float* ar_wd0 = (const float*)d_in[8];
    const float* b_wd0  = (const float*)d_in[9];
    const float* W_ww1  = (const float*)d_in[10];
    const float* al_ww1 = (const float*)d_in[11];
    const float* ar_ww1 = (const float*)d_in[12];
    const float* b_ww1  = (const float*)d_in[13];
    const float* W_wd1  = (const float*)d_in[14];
    const float* al_wd1 = (const float*)d_in[15];
    const float* ar_wd1 = (const float*)d_in[16];
    const float* b_wd1  = (const float*)d_in[17];
    const int*   ww_src = (const int*)d_in[18];
    const int*   ww_dst = (const int*)d_in[19];
    const int*   wd_src = (const int*)d_in[20];
    const int*   wd_dst = (const int*)d_in[21];

    float* out_xw = (float*)d_out;                  // [NW,128]
    float* out_xd = out_xw + (size_t)NW * FOUT;     // [ND,128]

    // workspace carve-out (floats)
    float* ws = (float*)d_ws;
    size_t off = 0;
    Scratch S;
    S.Zs   = ws + off; off += (size_t)NW * NOUT;    // 40.96M
    S.Zd   = ws + off; off += (size_t)ND * NOUT;    //  8.19M
    S.el   = ws + off; off += (size_t)NW * HH;
    S.er   = ws + off; off += (size_t)NW * HH;
    S.m_u  = (unsigned*)(ws + off); off += (size_t)NW * HH;
    S.ssum = ws + off; off += (size_t)NW * HH;
    S.ebuf = ws + off; off += (size_t)EWW * HH;
    S.hw   = ws + off; off += (size_t)NW * FOUT;
    S.hd   = ws + off; off += (size_t)ND * FOUT;
    S.xw1  = ws + off; off += (size_t)NW * FOUT;
    S.xd1  = ws + off; off += (size_t)ND * FOUT;

    // ---------------- layer 0 (fin = 256) ----------------
    run_relation(x_word, x_word, 256, W_ww0, al_ww0, ar_ww0,
                 ww_src, ww_dst, NW, NW, EWW, S, S.hw, stream);
    run_relation(x_word, x_doc,  256, W_wd0, al_wd0, ar_wd0,
                 wd_src, wd_dst, NW, ND, EWD, S, S.hd, stream);
    bias_relu<<<ceil_div(NW * FOUT, 256), 256, 0, stream>>>(S.hw, b_ww0, S.xw1, NW);
    bias_relu<<<ceil_div(ND * FOUT, 256), 256, 0, stream>>>(S.hd, b_wd0, S.xd1, ND);

    // ---------------- layer 1 (fin = 128) ----------------
    run_relation(S.xw1, S.xw1, 128, W_ww1, al_ww1, ar_ww1,
                 ww_src, ww_dst, NW, NW, EWW, S, S.hw, stream);
    run_relation(S.xw1, S.xd1, 128, W_wd1, al_wd1, ar_wd1,
                 wd_src, wd_dst, NW, ND, EWD, S, S.hd, stream);
    bias_relu<<<ceil_div(NW * FOUT, 256), 256, 0, stream>>>(S.hw, b_ww1, out_xw, NW);
    bias_relu<<<ceil_div(ND * FOUT, 256), 256, 0, stream>>>(S.hd, b_wd1, out_xd, ND);
}